// OT_11098195493006
// MI455X (gfx1250) — compile-verified
//
#include <hip/hip_runtime.h>
#include <hip/hip_bf16.h>

typedef __attribute__((ext_vector_type(2))) float v2f;
typedef __attribute__((ext_vector_type(8))) float v8f;

#define WMMA_F32_4(a, b, c) \
  __builtin_amdgcn_wmma_f32_16x16x4_f32(false, (a), false, (b), (short)0, (c), false, false)

// ---------------------------------------------------------------------------
// GEMM (NT): C[m][n] = sum_k A[m][k] * W[n][k] + bias[n]
// One wave per 16x16 tile, V_WMMA_F32_16X16X4_F32, K stepped by 4.
// blockDim = 128 (4 waves); grid = (Ntiles/4, Mtiles).
// ---------------------------------------------------------------------------
__global__ void gemm_nt_wmma(const float* __restrict__ A, const float* __restrict__ W,
                             const float* __restrict__ bias, float* __restrict__ C,
                             int K, int lda, int ldw, int ldc)
{
  const int lane = threadIdx.x & 31;
  const int wave = threadIdx.x >> 5;
  const int l16  = lane & 15;
  const int half = lane >> 4;                 // 0: k={0,1}, 1: k={2,3} within each 4-step
  const int n0 = (blockIdx.x * 4 + wave) * 16;
  const int m0 = blockIdx.y * 16;

  const float* ap = A + (long)(m0 + l16) * lda + 2 * half;
  const float* wp = W + (long)(n0 + l16) * ldw + 2 * half;
  v8f acc = {};
  for (int k0 = 0; k0 < K; k0 += 4) {
    v2f a; a.x = ap[k0]; a.y = ap[k0 + 1];
    v2f b; b.x = wp[k0]; b.y = wp[k0 + 1];
    acc = WMMA_F32_4(a, b, acc);
  }
  const float bn = bias[n0 + l16];
#pragma unroll
  for (int r = 0; r < 8; ++r)
    C[(long)(m0 + r + 8 * half) * ldc + n0 + l16] = acc[r] + bn;
}

// ---------------------------------------------------------------------------
// sim[bh][i][j] = sum_c q[b][i][h][c] * k[b][j][h][c]   (beta_t == 1)
// blockDim=128; grid = (36/4=9, 16, 128).
// ---------------------------------------------------------------------------
__global__ void sim_wmma_kernel(const float* __restrict__ q, const float* __restrict__ k,
                                float* __restrict__ sim)
{
  const int bh = blockIdx.z;
  const int b = bh >> 3, h = bh & 7;
  const int lane = threadIdx.x & 31;
  const int wave = threadIdx.x >> 5;
  const int l16 = lane & 15, half = lane >> 4;
  const int j0 = (blockIdx.x * 4 + wave) * 16;
  const int i0 = blockIdx.y * 16;

  const float* ap = q + ((long)b * 256 + i0 + l16) * 512 + h * 64 + 2 * half;
  const float* wp = k + ((long)b * 576 + j0 + l16) * 512 + h * 64 + 2 * half;
  v8f acc = {};
#pragma unroll
  for (int k0 = 0; k0 < 64; k0 += 4) {
    v2f a; a.x = ap[k0]; a.y = ap[k0 + 1];
    v2f bb; bb.x = wp[k0]; bb.y = wp[k0 + 1];
    acc = WMMA_F32_4(a, bb, acc);
  }
  float* C = sim + (long)bh * 256 * 576;
#pragma unroll
  for (int r = 0; r < 8; ++r)
    C[(long)(i0 + r + 8 * half) * 576 + j0 + l16] = acc[r];
}

// ---------------------------------------------------------------------------
// att[b][i][h*64+c] = sum_j score[bh][i][j] * v[b][j][h][c]    (B is K-major)
// blockDim=128 (4 waves == 4 c-tiles); grid = (1, 16, 128).
// ---------------------------------------------------------------------------
__global__ void av_wmma_kernel(const float* __restrict__ score, const float* __restrict__ v,
                               float* __restrict__ out)
{
  const int bh = blockIdx.z;
  const int b = bh >> 3, h = bh & 7;
  const int lane = threadIdx.x & 31;
  const int wave = threadIdx.x >> 5;
  const int l16 = lane & 15, half = lane >> 4;
  const int c0 = wave * 16;
  const int i0 = blockIdx.y * 16;

  const float* ap = score + ((long)bh * 256 + i0 + l16) * 576 + 2 * half;
  const float* B  = v + (long)b * 576 * 512 + h * 64 + c0 + l16;
  v8f acc = {};
  for (int k0 = 0; k0 < 576; k0 += 4) {
    v2f a; a.x = ap[k0]; a.y = ap[k0 + 1];
    const float* bp = B + (long)(k0 + 2 * half) * 512;
    v2f bb; bb.x = bp[0]; bb.y = bp[512];
    acc = WMMA_F32_4(a, bb, acc);
  }
  float* C = out + ((long)b * 256 + i0) * 512 + h * 64 + c0;
#pragma unroll
  for (int r = 0; r < 8; ++r)
    C[(long)(r + 8 * half) * 512 + l16] = acc[r];
}

// ---------------------------------------------------------------------------
// Per-64-element L2 norm with 1e-12 clamp (q and k). One wave per segment.
// ---------------------------------------------------------------------------
__global__ void l2norm64_kernel(float* __restrict__ x, int nseg)
{
  const int seg = blockIdx.x * 8 + (int)(threadIdx.x >> 5);
  if (seg >= nseg) return;
  const int lane = threadIdx.x & 31;
  float* p = x + (long)seg * 64;
  float a = p[lane], b = p[lane + 32];
  float ss = a * a + b * b;
#pragma unroll
  for (int off = 16; off > 0; off >>= 1) ss += __shfl_xor(ss, off, 32);
  const float inv = 1.0f / fmaxf(sqrtf(ss), 1e-12f);
  p[lane] = a * inv;
  p[lane + 32] = b * inv;
}

// ---------------------------------------------------------------------------
// Persistent Sinkhorn: 1 workgroup (1024 thr = 32 waves) per bh matrix.
// 100 Jacobi iterations; u/v/colsum in LDS; one streaming read of the
// 256x576 slab per iteration (L2-resident after iter 0).
// Row i is owned entirely by one wave -> rowsum via __shfl_xor.
// colsum via ds_add_f32 LDS atomics (distinct addresses per lane).
// ---------------------------------------------------------------------------
__global__ void sinkhorn_kernel(const float* __restrict__ sim,
                                float* __restrict__ u_g, float* __restrict__ v_g)
{
  __shared__ float u[256];
  __shared__ float v[576];
  __shared__ float colsum[576];

  const int bh = blockIdx.x;
  const float* S = sim + (long)bh * 256 * 576;
  const int t = threadIdx.x;
  const int lane = t & 31;
  const int wave = t >> 5;                    // 0..31, owns rows wave*8 .. wave*8+7

  for (int i = t; i < 256; i += 1024) u[i] = 0.0f;
  for (int j = t; j < 576; j += 1024) { v[j] = 0.0f; colsum[j] = 0.0f; }
  __syncthreads();

  const float eps = 0.05f, inv_eps = 20.0f;
  const float log_mu = __logf(1.0f / 256.0f + 1e-8f);
  const float log_nu = __logf(1.0f / 576.0f + 1e-8f);

  for (int it = 0; it < 100; ++it) {
    float unew[8];
#pragma unroll
    for (int r = 0; r < 8; ++r) {
      const int i = wave * 8 + r;
      const float ui = u[i];
      const float* Srow = S + (long)i * 576;
      float racc = 0.0f;
#pragma unroll
      for (int tc = 0; tc < 18; ++tc) {
        const int j = lane + tc * 32;
        const float s = Srow[j];                       // C = 1 - s
        const float e = __expf((ui + v[j] - (1.0f - s)) * inv_eps);
        racc += e;
        atomicAdd(&colsum[j], e);                      // ds_add_f32
      }
#pragma unroll
      for (int off = 16; off > 0; off >>= 1) racc += __shfl_xor(racc, off, 32);
      unew[r] = eps * (log_mu - __logf(racc)) + ui;
    }
    // only this wave ever reads u[wave*8..+7] within an iteration -> safe in-place
    if (lane == 0) {
#pragma unroll
      for (int r = 0; r < 8; ++r) u[wave * 8 + r] = unew[r];
    }
    __syncthreads();                                   // colsum complete
    for (int j = t; j < 576; j += 1024) {
      v[j] = eps * (log_nu - __logf(colsum[j])) + v[j];
      colsum[j] = 0.0f;
    }
    __syncthreads();
  }

  for (int i = t; i < 256; i += 1024) u_g[bh * 256 + i] = u[i];
  for (int j = t; j < 576; j += 1024) v_g[bh * 576 + j] = v[j];
}

// ---------------------------------------------------------------------------
// In-place: sim <- N1*N2 * sim * exp((u_i + v_j - (1 - sim)) / eps)
// ---------------------------------------------------------------------------
__global__ void score_kernel(float* __restrict__ s,
                             const float* __restrict__ ug, const float* __restrict__ vg)
{
  const long idx = (long)blockIdx.x * blockDim.x + threadIdx.x;   // exactly 128*256*576
  const int bh = (int)(idx / (256 * 576));
  const int rem = (int)(idx - (long)bh * (256 * 576));
  const int i = rem / 576, j = rem - i * 576;
  const float sv = s[idx];
  const float T = __expf((ug[bh * 256 + i] + vg[bh * 576 + j] - (1.0f - sv)) * 20.0f);
  s[idx] = 147456.0f * sv * T;                                    // N1*N2 = 256*576
}

// ---------------------------------------------------------------------------
// Final row L2 normalization over 512 (plain division, no clamp). Wave/row.
// ---------------------------------------------------------------------------
__global__ void rownorm512_kernel(const float* __restrict__ x, float* __restrict__ out, int rows)
{
  const int row = blockIdx.x * 8 + (int)(threadIdx.x >> 5);
  if (row >= rows) return;
  const int lane = threadIdx.x & 31;
  const float* p = x + (long)row * 512;
  float vals[16];
  float ss = 0.0f;
#pragma unroll
  for (int tcb = 0; tcb < 16; ++tcb) { vals[tcb] = p[lane + 32 * tcb]; ss += vals[tcb] * vals[tcb]; }
#pragma unroll
  for (int off = 16; off > 0; off >>= 1) ss += __shfl_xor(ss, off, 32);
  const float inv = 1.0f / sqrtf(ss);
  float* o = out + (long)row * 512;
#pragma unroll
  for (int tcb = 0; tcb < 16; ++tcb) o[lane + 32 * tcb] = vals[tcb] * inv;
}

// ---------------------------------------------------------------------------
extern "C" void kernel_launch(void* const* d_in, const int* in_sizes, int n_in,
                              void* d_out, int out_size, void* d_ws, size_t ws_size,
                              hipStream_t stream)
{
  const float* F_t = (const float*)d_in[0];   // [16,256,512]
  const float* F_s = (const float*)d_in[1];   // [16,576,512]
  const float* Wq  = (const float*)d_in[2];
  const float* bq  = (const float*)d_in[3];
  const float* Wk  = (const float*)d_in[4];
  const float* bk  = (const float*)d_in[5];
  const float* Wv  = (const float*)d_in[6];
  const float* bv  = (const float*)d_in[7];
  const float* Wp  = (const float*)d_in[8];
  const float* bp  = (const float*)d_in[9];
  float* out = (float*)d_out;

  float* ws   = (float*)d_ws;
  size_t o = 0;
  float* q    = ws + o; o += 2097152;      // 16*256*512
  float* kk   = ws + o; o += 4718592;      // 16*576*512
  float* vv   = ws + o; o += 4718592;      // 16*576*512
  float* simb = ws + o; o += 18874368;     // 128*256*576
  float* ug   = ws + o; o += 32768;        // 128*256
  float* vg   = ws + o; o += 73728;        // 128*576
  float* att  = ws + o; o += 2097152;      // 16*256*512
  float* proj = q;                          // q dead after sim -> reuse

  const dim3 blk(128);

  // projections: out = F @ W^T + b
  gemm_nt_wmma<<<dim3(8, 256, 1), blk, 0, stream>>>(F_t, Wq, bq, q,  512, 512, 512, 512);
  gemm_nt_wmma<<<dim3(8, 576, 1), blk, 0, stream>>>(F_s, Wk, bk, kk, 512, 512, 512, 512);
  gemm_nt_wmma<<<dim3(8, 576, 1), blk, 0, stream>>>(F_s, Wv, bv, vv, 512, 512, 512, 512);

  // per-head l2 normalization of q, k
  l2norm64_kernel<<<dim3(4096), dim3(256), 0, stream>>>(q,  32768);
  l2norm64_kernel<<<dim3(9216), dim3(256), 0, stream>>>(kk, 73728);

  // cosine similarity per (b,h)
  sim_wmma_kernel<<<dim3(9, 16, 128), blk, 0, stream>>>(q, kk, simb);

  // persistent 100-iteration Sinkhorn, 1 WG per bh
  sinkhorn_kernel<<<dim3(128), dim3(1024), 0, stream>>>(simb, ug, vg);

  // score map in place over sim
  score_kernel<<<dim3(73728), dim3(256), 0, stream>>>(simb, ug, vg);

  // score @ v
  av_wmma_kernel<<<dim3(1, 16, 128), blk, 0, stream>>>(simb, vv, att);

  // final projection + row norm
  gemm_nt_wmma<<<dim3(8, 256, 1), blk, 0, stream>>>(att, Wp, bp, proj, 512, 512, 512, 512);
  rownorm512_kernel<<<dim3(512), dim3(256), 0, stream>>>(proj, out, 4096);
}